// MILClassifier_44633300140138
// MI455X (gfx1250) — compile-verified
//
#include <hip/hip_runtime.h>
#include <stdint.h>

#define Bb   64
#define Nn   2048
#define Dd   512
#define Cc   128
#define TK   8
#define LN_EPS 1e-5f
#define APAD 516  // LDS row stride = D + 4 floats -> bank-conflict-free column reads

typedef float v2f __attribute__((ext_vector_type(2)));
typedef float v8f __attribute__((ext_vector_type(8)));

__device__ __forceinline__ float wave_sum(float v) {
#pragma unroll
  for (int off = 16; off > 0; off >>= 1) v += __shfl_xor(v, off, 32);
  return v;
}

// ---------------- Kernel 1: fused LayerNorm-stats + score GEMV ----------------
// grid = B * N/8 blocks of 256 (8 wave32s, one clip per wave). Single pass over x.
__global__ __launch_bounds__(256) void mil_score_kernel(
    const float* __restrict__ x, const float* __restrict__ ln_w,
    const float* __restrict__ /*ln_b: constant shift, drops out of ranking*/,
    const float* __restrict__ W, const uint8_t* __restrict__ mask,
    const int* __restrict__ y, float* __restrict__ score) {
  __shared__ __align__(16) float gsh[Dd];
  const int b   = blockIdx.x >> 8;          // 256 blocks per batch
  const int n0  = (blockIdx.x & 255) * 8;
  const int tid = threadIdx.x;
  const int yb  = y[b];
  for (int d = tid; d < Dd; d += 256)
    gsh[d] = ln_w[d] * W[(size_t)d * Cc + yb];
  __syncthreads();

  const int wave = tid >> 5, lane = tid & 31;
  const int n = n0 + wave;
  const float* xr = x + ((size_t)b * Nn + n) * Dd;
  float sx = 0.f, sxx = 0.f, sxg = 0.f, sg = 0.f;
#pragma unroll
  for (int i = 0; i < 4; ++i) {
    const int d = i * 128 + lane * 4;
    const float4 v = *(const float4*)(xr + d);
    const float4 g = *(const float4*)(gsh + d);
    sx  += v.x + v.y + v.z + v.w;
    sxx += v.x * v.x + v.y * v.y + v.z * v.z + v.w * v.w;
    sxg += v.x * g.x + v.y * g.y + v.z * g.z + v.w * g.w;
    sg  += g.x + g.y + g.z + g.w;
  }
  sx = wave_sum(sx); sxx = wave_sum(sxx); sxg = wave_sum(sxg); sg = wave_sum(sg);
  if (lane == 0) {
    const float inv = 1.f / (float)Dd;
    const float mu  = sx * inv;
    const float rs  = rsqrtf(sxx * inv - mu * mu + LN_EPS);
    const float core = rs * (sxg - mu * sg);   // batch-constant terms dropped
    score[(size_t)b * Nn + n] =
        mask[(size_t)b * Nn + n] ? core : -__builtin_inff();
  }
}

// -------- Kernel 2: per-batch top-8, re-LN selected rows, WMMA fp32 GEMM -----
__global__ __launch_bounds__(256) void mil_topk_wmma_kernel(
    const float* __restrict__ x, const float* __restrict__ ln_w,
    const float* __restrict__ ln_b, const float* __restrict__ W,
    const float* __restrict__ bias, const float* __restrict__ score,
    float* __restrict__ out) {
  __shared__ __align__(16) float axn[16 * APAD];
  __shared__ float red_s[256];
  __shared__ int   red_i[256];
  __shared__ int   sel[TK];
  const int b = blockIdx.x;
  const int tid = threadIdx.x;

  // --- phase 0: iterative top-8 argmax over score[b, :] (8 vals per thread) ---
  const float* sc = score + (size_t)b * Nn;
  float ls[8]; int li[8];
#pragma unroll
  for (int j = 0; j < 8; ++j) { const int i = j * 256 + tid; ls[j] = sc[i]; li[j] = i; }
  for (int t = 0; t < TK; ++t) {
    float best = -__builtin_inff(); int bi = -1;
#pragma unroll
    for (int j = 0; j < 8; ++j) if (ls[j] > best) { best = ls[j]; bi = li[j]; }
    red_s[tid] = best; red_i[tid] = bi;
    __syncthreads();
    for (int off = 128; off > 0; off >>= 1) {
      if (tid < off && red_s[tid + off] > red_s[tid]) {
        red_s[tid] = red_s[tid + off]; red_i[tid] = red_i[tid + off];
      }
      __syncthreads();
    }
    const int win = red_i[0];
    if (tid == 0) sel[t] = win;
#pragma unroll
    for (int j = 0; j < 8; ++j) if (li[j] == win) ls[j] = -__builtin_inff();
    __syncthreads();
  }

  // --- phase 1: LayerNorm the 8 selected rows into LDS rows 0..7; zero 8..15 ---
  const int wave = tid >> 5, lane = tid & 31;
  {
    const int n = sel[wave];
    const float* xr = x + ((size_t)b * Nn + n) * Dd;
    float4 v[4];
    float sx = 0.f, sxx = 0.f;
#pragma unroll
    for (int i = 0; i < 4; ++i) {
      const int d = i * 128 + lane * 4;
      v[i] = *(const float4*)(xr + d);
      sx  += v[i].x + v[i].y + v[i].z + v[i].w;
      sxx += v[i].x * v[i].x + v[i].y * v[i].y + v[i].z * v[i].z + v[i].w * v[i].w;
    }
    sx = wave_sum(sx); sxx = wave_sum(sxx);
    const float inv = 1.f / (float)Dd;
    const float mu = sx * inv;
    const float rs = rsqrtf(sxx * inv - mu * mu + LN_EPS);
#pragma unroll
    for (int i = 0; i < 4; ++i) {
      const int d = i * 128 + lane * 4;
      const float4 gw = *(const float4*)(ln_w + d);
      const float4 gb = *(const float4*)(ln_b + d);
      float* rowp = &axn[wave * APAD + d];
      rowp[0] = (v[i].x - mu) * rs * gw.x + gb.x;
      rowp[1] = (v[i].y - mu) * rs * gw.y + gb.y;
      rowp[2] = (v[i].z - mu) * rs * gw.z + gb.z;
      rowp[3] = (v[i].w - mu) * rs * gw.w + gb.w;
    }
  }
  for (int i = tid; i < 8 * APAD; i += 256) axn[8 * APAD + i] = 0.f;
  __syncthreads();

  // --- phase 2: xn(16x512) @ W(512x128) via v_wmma_f32_16x16x4_f32 ---
  // wave w owns output columns [16w, 16w+16). Fragment mapping (ISA 7.12.2):
  //   lane -> {h = lane>>4, r = lane&15}; component v holds K = kb + 2h + v;
  //   A row = r (from LDS, stride APAD), B col = r (from global W).
  const int c0 = wave * 16;
  const int r  = lane & 15;
  const int h  = lane >> 4;
  v8f acc = {};
  const float* wcol = W + c0 + r;
#pragma unroll 4
  for (int kb = 0; kb < Dd; kb += 4) {
    const int k = kb + 2 * h;
    const v2f a = *(const v2f*)(&axn[r * APAD + k]);
    v2f bm;
    bm.x = wcol[(size_t)k * Cc];
    bm.y = wcol[(size_t)(k + 1) * Cc];
    acc = __builtin_amdgcn_wmma_f32_16x16x4_f32(
        /*neg_a=*/false, a, /*neg_b=*/false, bm,
        /*c_mod=*/(short)0, acc, /*reuse_a=*/false, /*reuse_b=*/false);
  }
  // Rows 0..7 of D live in acc[0..7] on lanes 0..15 (rows 8..15 are the zero pad).
  const float s = acc[0] + acc[1] + acc[2] + acc[3] + acc[4] + acc[5] + acc[6] + acc[7];
  if (lane < 16) out[b * Cc + c0 + lane] = s * 0.125f + bias[c0 + lane];
}

extern "C" void kernel_launch(void* const* d_in, const int* in_sizes, int n_in,
                              void* d_out, int out_size, void* d_ws, size_t ws_size,
                              hipStream_t stream) {
  const float*   x    = (const float*)d_in[0];
  const float*   ln_w = (const float*)d_in[1];
  const float*   ln_b = (const float*)d_in[2];
  const float*   W    = (const float*)d_in[3];
  const float*   bias = (const float*)d_in[4];
  const uint8_t* mask = (const uint8_t*)d_in[5];
  const int*     y    = (const int*)d_in[6];
  float* out   = (float*)d_out;
  float* score = (float*)d_ws;   // B*N floats = 512 KB

  mil_score_kernel<<<dim3(Bb * (Nn / 8)), 256, 0, stream>>>(
      x, ln_w, ln_b, W, mask, y, score);
  mil_topk_wmma_kernel<<<dim3(Bb), 256, 0, stream>>>(
      x, ln_w, ln_b, W, bias, score, out);
}